// GATLayer_39719857553790
// MI455X (gfx1250) — compile-verified
//
#include <hip/hip_runtime.h>
#include <hip/hip_bf16.h>
#include <math.h>

#define NN   8192      // nodes
#define NE   262144    // edges
#define NH   8         // heads
#define DP   32        // d' per head
#define DIN  256       // input dim
#define DCAT 256       // H * DP (concat output dim)
#define MAXDEG 1024    // LDS sort capacity per row (max observed deg ~60)

typedef float v2f __attribute__((ext_vector_type(2)));
typedef float v8f __attribute__((ext_vector_type(8)));

// ---------------------------------------------------------------------------
// Wx = x @ Wcat^T + bias  :  [8192 x 256] = [8192 x 256] * [256 x 256]^T
// fp32 WMMA 16x16x4, one wave per 16x16 output tile.
// A frag (16x4 f32): lane m = lane&15, khalf = lane>>4; VGPR0/1 = K(2*khalf)/(2*khalf+1)
// B frag (4x16 f32): lane n = lane&15, same K mapping (B[k][n] = Wcat[n][k])
// C/D (16x16 f32):   D[v + 8*khalf][lane&15] = acc[v]
// ---------------------------------------------------------------------------
__global__ void gemm_wmma_f32(const float* __restrict__ x,
                              const float* __restrict__ W,
                              const float* __restrict__ bias,
                              float* __restrict__ Wx) {
  int wave  = (blockIdx.x * blockDim.x + threadIdx.x) >> 5;
  int lane  = threadIdx.x & 31;
  int mtile = wave >> 4;       // 0..511
  int ntile = wave & 15;       // 0..15
  int mi    = lane & 15;
  int kh    = lane >> 4;       // 0 or 1

  const float* arow = x + (size_t)(mtile * 16 + mi) * DIN + 2 * kh;
  const float* brow = W + (size_t)(ntile * 16 + mi) * DIN + 2 * kh;

  v8f acc = {};
  for (int kb = 0; kb < DIN; kb += 4) {
    v2f a = *(const v2f*)(arow + kb);
    v2f b = *(const v2f*)(brow + kb);
    acc = __builtin_amdgcn_wmma_f32_16x16x4_f32(
        /*neg_a=*/false, a, /*neg_b=*/false, b,
        /*c_mod=*/(short)0, acc, /*reuse_a=*/false, /*reuse_b=*/false);
  }

  int col = ntile * 16 + mi;
  float bv = bias[col];
#pragma unroll
  for (int v = 0; v < 8; ++v) {
    int row = mtile * 16 + v + 8 * kh;
    Wx[(size_t)row * DCAT + col] = acc[v] + bv;
  }
}

// ---------------------------------------------------------------------------
// sl[k][i] = Wx_k[i] . a_w[k][0:32] ; sr[k][i] = Wx_k[i] . a_w[k][32:64]
// one wave per (node, head): lanes over d', shuffle reduce.
// ---------------------------------------------------------------------------
__global__ void attn_proj(const float* __restrict__ Wx,
                          const float* __restrict__ aw,
                          float* __restrict__ sl, float* __restrict__ sr) {
  int i = blockIdx.x;
  int k = threadIdx.x >> 5;
  int d = threadIdx.x & 31;
  float v  = Wx[(size_t)i * DCAT + k * DP + d];
  float p1 = v * aw[k * 2 * DP + d];
  float p2 = v * aw[k * 2 * DP + DP + d];
#pragma unroll
  for (int off = 16; off > 0; off >>= 1) {
    p1 += __shfl_down(p1, off, 32);
    p2 += __shfl_down(p2, off, 32);
  }
  if (d == 0) {
    sl[k * NN + i] = p1;
    sr[k * NN + i] = p2;
  }
}

// S[c] = sum_i Wx[i][c]   (256 columns, one block, coalesced row sweeps)
__global__ void colsum(const float* __restrict__ Wx, float* __restrict__ S) {
  int t = threadIdx.x;
  float s = 0.f;
  for (int i = 0; i < NN; ++i) s += Wx[(size_t)i * DCAT + t];
  S[t] = s;
}

// ------------------------- CSR construction -------------------------------
__global__ void edge_count(const int* __restrict__ ei, int* __restrict__ rowcnt) {
  int e = blockIdx.x * blockDim.x + threadIdx.x;
  if (e < NE) atomicAdd(&rowcnt[ei[e]], 1);
}

__global__ void scan_rows(const int* __restrict__ rowcnt,
                          int* __restrict__ rowptr, int* __restrict__ rowpos) {
  int s = 0;
  for (int i = 0; i < NN; ++i) {
    rowptr[i] = s;
    rowpos[i] = s;
    s += rowcnt[i];
  }
  rowptr[NN] = s;
}

__global__ void edge_scatter(const int* __restrict__ ei,
                             int* __restrict__ rowpos, int* __restrict__ colidx) {
  int e = blockIdx.x * blockDim.x + threadIdx.x;
  if (e < NE) {
    int vi = ei[e];
    int vj = ei[NE + e];
    int p = atomicAdd(&rowpos[vi], 1);
    colidx[p] = vj;
  }
}

// ---------------------------------------------------------------------------
// Per row: sort column list (deterministic order) + run-length multiplicity.
// mult[p] = m for first occurrence of a distinct column, 0 otherwise.
// One wave per row; insertion sort in LDS by lane 0 (deg ~32).
// ---------------------------------------------------------------------------
__global__ void row_sort_mult(const int* __restrict__ rowptr,
                              int* __restrict__ colidx, int* __restrict__ mult) {
  __shared__ int buf[MAXDEG];
  int row = blockIdx.x;
  int start = rowptr[row], end = rowptr[row + 1];
  int deg = end - start;
  int lane = threadIdx.x;
  if (deg <= 0) return;

  if (deg <= MAXDEG) {
    for (int p = lane; p < deg; p += 32) buf[p] = colidx[start + p];
    __syncthreads();
    if (lane == 0) {
      for (int p = 1; p < deg; ++p) {
        int key = buf[p];
        int q = p - 1;
        while (q >= 0 && buf[q] > key) { buf[q + 1] = buf[q]; --q; }
        buf[q + 1] = key;
      }
    }
    __syncthreads();
    for (int p = lane; p < deg; p += 32) {
      int j = buf[p];
      int m = 0;
      if (p == 0 || buf[p - 1] != j) {
        int q = p;
        while (q < deg && buf[q] == j) { ++m; ++q; }
      }
      colidx[start + p] = j;
      mult[start + p] = m;
    }
  } else {  // safety fallback, O(deg^2) from global (statistically unreachable)
    for (int p = lane; p < deg; p += 32) {
      int j = colidx[start + p];
      bool first = true;
      for (int q = 0; q < p; ++q)
        if (colidx[start + q] == j) { first = false; break; }
      int m = 0;
      if (first)
        for (int q = p; q < deg; ++q)
          if (colidx[start + q] == j) ++m;
      mult[start + p] = m;
    }
  }
}

// ---------------------------------------------------------------------------
// out[i, k*32+d] = elu( (S_k[d] + sum_edges (exp(m*e)-1)*Wx_k[j][d])
//                       / (N + sum_edges (exp(m*e)-1)) )
// wave per (row, head); lane = d.
// ---------------------------------------------------------------------------
__global__ void aggregate(const float* __restrict__ Wx,
                          const float* __restrict__ sl, const float* __restrict__ sr,
                          const float* __restrict__ ab, const float* __restrict__ S,
                          const int* __restrict__ rowptr, const int* __restrict__ colidx,
                          const int* __restrict__ mult, float* __restrict__ out) {
  int row = blockIdx.x;
  int k = threadIdx.x >> 5;
  int d = threadIdx.x & 31;
  int start = rowptr[row], end = rowptr[row + 1];

  float slv = sl[k * NN + row];
  float abv = ab[k];
  const float* srk = sr + k * NN;

  float denom = 0.f, acc = 0.f;
  for (int p = start; p < end; ++p) {
    int m = mult[p];
    if (m == 0) continue;  // duplicate, already folded into first occurrence
    int j = colidx[p];
    float t = slv + srk[j] + abv;
    float e = t > 0.f ? t : 0.2f * t;       // leaky_relu(0.2)
    float w = expf((float)m * e) - 1.0f;    // exp(A[i,j]) - exp(0)
    denom += w;
    acc += w * Wx[(size_t)j * DCAT + k * DP + d];
  }
  float val = (S[k * DP + d] + acc) / ((float)NN + denom);
  out[(size_t)row * DCAT + k * DP + d] = val > 0.f ? val : expm1f(val);  // elu
}

// ---------------------------------------------------------------------------
extern "C" void kernel_launch(void* const* d_in, const int* in_sizes, int n_in,
                              void* d_out, int out_size, void* d_ws, size_t ws_size,
                              hipStream_t stream) {
  const float* x  = (const float*)d_in[0];
  const int*   ei = (const int*)d_in[1];   // [2, E] int32 (JAX default x64 off)
  const float* Ww = (const float*)d_in[2]; // [8,32,256] == Wcat [256,256]
  const float* Wb = (const float*)d_in[3]; // [8,32]    == flat [256]
  const float* aw = (const float*)d_in[4]; // [8,1,64]
  const float* ab = (const float*)d_in[5]; // [8,1]
  float* out = (float*)d_out;

  char* ws = (char*)d_ws;
  size_t off = 0;
  auto alloc = [&](size_t bytes) -> void* {
    void* p = ws + off;
    off = (off + bytes + 255) & ~(size_t)255;
    return p;
  };
  float* Wx     = (float*)alloc((size_t)NN * DCAT * sizeof(float)); // 8 MB
  float* sl     = (float*)alloc((size_t)NH * NN * sizeof(float));
  float* sr     = (float*)alloc((size_t)NH * NN * sizeof(float));
  float* S      = (float*)alloc(DCAT * sizeof(float));
  int*   rowcnt = (int*)alloc(NN * sizeof(int));
  int*   rowptr = (int*)alloc((NN + 1) * sizeof(int));
  int*   rowpos = (int*)alloc(NN * sizeof(int));
  int*   colidx = (int*)alloc((size_t)NE * sizeof(int));
  int*   multb  = (int*)alloc((size_t)NE * sizeof(int));
  (void)in_sizes; (void)n_in; (void)out_size; (void)ws_size;

  hipMemsetAsync(rowcnt, 0, NN * sizeof(int), stream);

  gemm_wmma_f32<<<(NN / 16) * (DCAT / 16) / 8, 256, 0, stream>>>(x, Ww, Wb, Wx);
  attn_proj<<<NN, 256, 0, stream>>>(Wx, aw, sl, sr);
  colsum<<<1, 256, 0, stream>>>(Wx, S);

  edge_count<<<NE / 256, 256, 0, stream>>>(ei, rowcnt);
  scan_rows<<<1, 1, 0, stream>>>(rowcnt, rowptr, rowpos);
  edge_scatter<<<NE / 256, 256, 0, stream>>>(ei, rowpos, colidx);
  row_sort_mult<<<NN, 32, 0, stream>>>(rowptr, colidx, multb);

  aggregate<<<NN, 256, 0, stream>>>(Wx, sl, sr, ab, S, rowptr, colidx, multb, out);
}